// MultiBoxLoss_1486058684466
// MI455X (gfx1250) — compile-verified
//
#include <hip/hip_runtime.h>

// ---------------------------------------------------------------------------
// SSD MultiBox loss for MI455X (gfx1250, wave32).
//   B=128 rows, A=8732 anchors, C=21 classes.
//   K1: stream confidences (94 MB) via async global->LDS b128 copies
//       (double-buffered, ASYNCcnt-pipelined), CE per anchor, partials.
//   K2: per-row exact top-k (radix select in LDS) of negative CE; row data
//       staged into LDS with async copies.
//   K3: deterministic final reduce + normalize.
// Reductions use V_WMMA_F32_16X16X4_F32 (f32-exact, deterministic).
// ---------------------------------------------------------------------------

#define NB 128
#define NA 8732
#define NC 21
#define SPLIT 8
#define CHUNK ((NA + SPLIT - 1) / SPLIT) // 1092
#define T1 256
#define T2 256
#define TILE 256 // anchors per K1 tile; 256*21*4 = 21504 B (16B aligned)

typedef float v2f __attribute__((ext_vector_type(2)));
typedef float v8f __attribute__((ext_vector_type(8)));

// --------------------------- WMMA wave reduction ---------------------------
// A[m][k]: lane L holds A[L%16][2*(L/16)+j] in vgpr j -> v in j=0, 0 in j=1.
// B = all ones (4x16) => D[m][n] = v_m + v_{m+16} (f32 exact, fixed order).
// Sum D's 8 vgprs per lane (half-row sum), add other half via shfl_xor(16).
__device__ __forceinline__ float wave_reduce_sum_wmma(float v) {
  v2f a; a[0] = v; a[1] = 0.0f;
  v2f ones; ones[0] = 1.0f; ones[1] = 1.0f;
  v8f c = {0.f, 0.f, 0.f, 0.f, 0.f, 0.f, 0.f, 0.f};
  v8f d = __builtin_amdgcn_wmma_f32_16x16x4_f32(
      false, a, false, ones, (short)0, c, false, false);
  float s = ((d[0] + d[1]) + (d[2] + d[3])) + ((d[4] + d[5]) + (d[6] + d[7]));
  s += __shfl_xor(s, 16, 32);
  return s;
}

__device__ __forceinline__ float block_reduce_sum_256(float v, float* s8) {
  float w = wave_reduce_sum_wmma(v);
  if ((threadIdx.x & 31u) == 0u) s8[threadIdx.x >> 5] = w;
  __syncthreads();
  float r = ((s8[0] + s8[1]) + (s8[2] + s8[3])) + ((s8[4] + s8[5]) + (s8[6] + s8[7]));
  __syncthreads();
  return r;
}

// --------------------------- async copy helpers ----------------------------
// Generic pointers to LDS are {aperture_hi32, lds_offset_lo32}; low 32 bits
// are exactly the byte address the async op's VDST expects.
__device__ __forceinline__ unsigned lds_addr_of(const void* p) {
  return (unsigned)(size_t)p;
}

// Copy nbytes (multiple of 16) from gbase to LDS at lds_base using b128 async
// loads. All lanes stay active (clamped chunk ids duplicate the tail chunk),
// so every wave issues exactly `nj` async instructions. Returns nj.
__device__ __forceinline__ int async_copy_to_lds(unsigned lds_base,
                                                 unsigned long long gbase,
                                                 int nbytes, int tid, int nthr) {
  const int nchunks = nbytes >> 4;
  const int nj = (nchunks + nthr - 1) / nthr;
  for (int j = 0; j < nj; ++j) {
    int c = j * nthr + tid;
    if (c >= nchunks) c = nchunks - 1;
    const unsigned off = (unsigned)(c << 4);
    const unsigned loff = lds_base + off;
    asm volatile("global_load_async_to_lds_b128 %0, %1, %2"
                 :: "v"(loff), "v"(off), "s"(gbase) : "memory");
  }
  return nj;
}

// Wait until outstanding async ops <= n (n is workgroup-uniform).
__device__ __forceinline__ void wait_async_le(int n) {
  if (n <= 0)      asm volatile("s_wait_asynccnt 0x0" ::: "memory");
  else if (n == 1) asm volatile("s_wait_asynccnt 0x1" ::: "memory");
  else if (n == 2) asm volatile("s_wait_asynccnt 0x2" ::: "memory");
  else if (n == 3) asm volatile("s_wait_asynccnt 0x3" ::: "memory");
  else if (n == 4) asm volatile("s_wait_asynccnt 0x4" ::: "memory");
  else if (n == 5) asm volatile("s_wait_asynccnt 0x5" ::: "memory");
  else             asm volatile("s_wait_asynccnt 0x6" ::: "memory");
}

// ---------------------------------------------------------------------------
// Kernel 1: grid (SPLIT, NB). CE per anchor + positive-anchor partials.
// Confidences staged tile-by-tile into double-buffered LDS via async copies.
// ---------------------------------------------------------------------------
__global__ __launch_bounds__(T1) void k1_ce(
    const float* __restrict__ conf, const float* __restrict__ loc,
    const int* __restrict__ lab, const float* __restrict__ gtl,
    float* __restrict__ ce_out, float* __restrict__ partials) {
  __shared__ float s8[8];
  __shared__ float sconf[2][TILE * NC]; // 2 x 21504 B

  const int b = blockIdx.y;
  const int s = blockIdx.x;
  const int a0 = s * CHUNK;
  const int a1 = (a0 + CHUNK < NA) ? (a0 + CHUNK) : NA;
  const int tid = (int)threadIdx.x;
  const int nt = (a1 - a0 + TILE - 1) / TILE; // == 5 for all blocks

  const unsigned long long rowbase =
      (unsigned long long)(const void*)conf +
      (unsigned long long)((size_t)b * NA * NC * 4);

  float nposf = 0.f, cepos = 0.f, locs = 0.f;

  // Issue the async copy for tile t into buffer t&1; returns #ops per wave.
  auto issue = [&](int t) -> int {
    const int ts = a0 + t * TILE;
    int len = a1 - ts; if (len > TILE) len = TILE; // 256, 68 or 64 (mult of 4)
    const unsigned long long gbase = rowbase + (unsigned long long)ts * (NC * 4);
    const unsigned lbase = lds_addr_of(&sconf[t & 1][0]);
    return async_copy_to_lds(lbase, gbase, len * NC * 4, tid, T1);
  };

  issue(0);
  for (int t = 0; t < nt; ++t) {
    int njn = 0;
    if (t + 1 < nt) njn = issue(t + 1);
    wait_async_le(njn);   // async loads retire in order -> tile t landed
    __syncthreads();      // all waves' copies visible

    const float* cbuf = &sconf[t & 1][0];
    const int a = a0 + t * TILE + tid;
    if (a < a1) {
      const size_t base = (size_t)b * NA + (size_t)a;
      const int lb = lab[base];

      float xs[NC];
      float m = -3.402823466e+38f;
#pragma unroll
      for (int i = 0; i < NC; ++i) {
        xs[i] = cbuf[tid * NC + i]; // stride 21 dwords: conflict-free (21 || 64)
        m = fmaxf(m, xs[i]);
      }
      float se = 0.f, xl = 0.f;
#pragma unroll
      for (int i = 0; i < NC; ++i) {
        se += __expf(xs[i] - m);
        xl = (i == lb) ? xs[i] : xl; // register select, no scratch
      }
      const float ce = m + __logf(se) - xl; // -log_softmax[label] >= 0

      float cen;
      if (lb > 0) {
        nposf += 1.f;
        cepos += ce;
        const float* lp = loc + base * 4;
        const float* gp = gtl + base * 4;
#pragma unroll
        for (int j = 0; j < 4; ++j) {
          const float dd = lp[j] - gp[j];
          const float ad = fabsf(dd);
          locs += (ad < 1.f) ? 0.5f * dd * dd : (ad - 0.5f);
        }
        cen = 0.f;
      } else {
        cen = ce;
      }
      ce_out[base] = cen;
    }
    __syncthreads(); // done reading buf[t&1] before it is overwritten (t+2)
  }

  const float rn = block_reduce_sum_256(nposf, s8);
  const float rc = block_reduce_sum_256(cepos, s8);
  const float rl = block_reduce_sum_256(locs, s8);
  if (tid == 0) {
    float* p = partials + ((size_t)b * SPLIT + s) * 3;
    p[0] = rn; p[1] = rc; p[2] = rl;
  }
}

// ---------------------------------------------------------------------------
// Kernel 2: grid (NB). Row top-k sum of negative CE via radix select in LDS.
// Nonnegative floats: IEEE bit pattern order == value order.
// ---------------------------------------------------------------------------
__global__ __launch_bounds__(T2) void k2_select(
    const float* __restrict__ ce_in, const float* __restrict__ partials,
    float* __restrict__ row_loss, float* __restrict__ row_np) {
  __shared__ unsigned int sbits[NA];
  __shared__ unsigned int hist[256];
  __shared__ float s8[8];
  __shared__ float sred[3];
  __shared__ int sk;
  __shared__ unsigned int ssel;
  __shared__ unsigned int scnt;

  const int b = blockIdx.x;
  const int tid = (int)threadIdx.x;

  // Stage the whole row's CE bits straight into LDS (34928 B, 16B aligned).
  {
    const unsigned long long gbase =
        (unsigned long long)(const void*)(ce_in + (size_t)b * NA);
    async_copy_to_lds(lds_addr_of(&sbits[0]), gbase, NA * 4, tid, T2);
  }

  if (tid == 0) {
    float n = 0.f, c = 0.f, l = 0.f;
    for (int s = 0; s < SPLIT; ++s) {
      const float* p = partials + ((size_t)b * SPLIT + s) * 3;
      n += p[0]; c += p[1]; l += p[2];
    }
    sred[0] = n; sred[1] = c; sred[2] = l;
    const int np = (int)(n + 0.5f);
    int k = 3 * np;
    const int negav = NA - np;
    if (k > negav) k = negav;
    sk = k;
  }
  wait_async_le(0);
  __syncthreads();

  const int k = sk; // workgroup-uniform
  float sum_top = 0.f;
  if (k > 0) {
    unsigned int prefix = 0u, maskv = 0u;
    unsigned int rem = (unsigned int)k;
    for (int shift = 24; shift >= 0; shift -= 8) {
      hist[tid] = 0u; // T2 == 256
      __syncthreads();
      for (int a = tid; a < NA; a += T2) {
        const unsigned int v = sbits[a];
        if ((v & maskv) == prefix) atomicAdd(&hist[(v >> shift) & 255u], 1u);
      }
      __syncthreads();
      if (tid == 0) {
        unsigned int cum = 0u;
        int dig = 255;
        for (; dig > 0; --dig) {
          const unsigned int h = hist[dig];
          if (cum + h >= rem) break;
          cum += h;
        }
        ssel = (unsigned int)dig;
        scnt = cum;
      }
      __syncthreads();
      prefix |= ssel << shift;
      rem -= scnt;
      maskv |= 0xFFu << shift;
      __syncthreads();
    }
    // prefix = k-th largest value's bits; rem = #pivot-equal elements to take
    float part = 0.f;
    for (int a = tid; a < NA; a += T2) {
      const unsigned int v = sbits[a];
      if (v > prefix) part += __uint_as_float(v);
    }
    sum_top = block_reduce_sum_256(part, s8);
    sum_top += (float)rem * __uint_as_float(prefix);
  }

  if (tid == 0) {
    row_loss[b] = sred[2] + sred[1] + sum_top; // loc + pos CE + mined neg CE
    row_np[b] = sred[0];
  }
}

// ---------------------------------------------------------------------------
// Kernel 3: 1 block, NB threads. Deterministic tree reduce + normalize.
// ---------------------------------------------------------------------------
__global__ __launch_bounds__(NB) void k3_final(
    const float* __restrict__ row_loss, const float* __restrict__ row_np,
    float* __restrict__ out) {
  __shared__ float sl[NB];
  __shared__ float sn[NB];
  const int t = (int)threadIdx.x;
  sl[t] = row_loss[t];
  sn[t] = row_np[t];
  __syncthreads();
  for (int off = NB / 2; off > 0; off >>= 1) {
    if (t < off) { sl[t] += sl[t + off]; sn[t] += sn[t + off]; }
    __syncthreads();
  }
  if (t == 0) out[0] = sl[0] / sn[0];
}

// ---------------------------------------------------------------------------
extern "C" void kernel_launch(void* const* d_in, const int* in_sizes, int n_in,
                              void* d_out, int out_size, void* d_ws, size_t ws_size,
                              hipStream_t stream) {
  const float* conf = (const float*)d_in[0]; // (B,A,C) f32
  const float* loc  = (const float*)d_in[1]; // (B,A,4) f32
  const int*   lab  = (const int*)d_in[2];   // (B,A)   i32
  const float* gtl  = (const float*)d_in[3]; // (B,A,4) f32
  float* out = (float*)d_out;

  float* ws = (float*)d_ws;
  float* ce       = ws;                                // NB*NA floats (~4.5 MB)
  float* partials = ce + (size_t)NB * NA;              // NB*SPLIT*3
  float* row_loss = partials + (size_t)NB * SPLIT * 3; // NB
  float* row_np   = row_loss + NB;                     // NB

  k1_ce<<<dim3(SPLIT, NB), T1, 0, stream>>>(conf, loc, lab, gtl, ce, partials);
  k2_select<<<dim3(NB), T2, 0, stream>>>(ce, partials, row_loss, row_np);
  k3_final<<<1, NB, 0, stream>>>(row_loss, row_np, out);
}